// FocalLossWithMask_22239340659435
// MI455X (gfx1250) — compile-verified
//
#include <hip/hip_runtime.h>
#include <hip/hip_bf16.h>

// ---------------------------------------------------------------------------
// Focal loss + hard negative mining, MI455X (gfx1250) version.
//   kernel 1: one block (1024 thr = 32 wave32) per row.
//             - single HBM streaming pass: compute focal loss, pack per-element
//               selection key into 320KB WGP LDS (row = 256KB, CDNA5-only fit)
//             - O(N) radix-select (3 x ds-atomic histogram + LDS suffix scan)
//               for the k-th largest negative loss (k = 3*num_pos)
//             - masked accumulation from LDS keys only (no 2nd global pass)
//   kernel 2: cross-row reduction of 128 (sum,count) pairs via
//             v_wmma_f32_16x16x4_f32 with A=ones, then wave32 shfl reduce.
// ---------------------------------------------------------------------------

typedef float v2f __attribute__((ext_vector_type(2)));
typedef float v8f __attribute__((ext_vector_type(8)));

#define N_COL  65536
#define NT     1024
#define NV4    (N_COL / 4)   // 128-bit chunks per row
#define NBINS  2048

// Per-element focal loss. Positives are accumulated immediately (always
// relevant). Negatives return a selection key:
//   bits30:0 = float bits of the UNWEIGHTED loss (loss >= 0 -> uint-ordered)
//   bit31    = fp flag (pred >= 0 -> final weight x3), ignored for ranking.
// key == 0 marks a positive.
__device__ __forceinline__ unsigned focal_elem(float x, float y,
                                               float& possum, int& poscnt)
{
    float ax  = fabsf(x);
    float l1p = __logf(1.0f + __expf(-ax));       // log1p(e^-|x|), arg in (1,2]
    float sp  = fmaxf(x, 0.0f) + l1p;             // softplus(x)
    float bce = y * (sp - x) + (1.0f - y) * sp;   // y*sp(-x) + (1-y)*sp(x)
    float p   = 1.0f / (1.0f + __expf(-x));       // sigmoid
    float pt  = y * p + (1.0f - y) * (1.0f - p);
    float om  = 1.0f - pt;
    float al  = y * 0.75f + (1.0f - y) * 0.25f;   // alpha weighting
    float loss = bce * om * om * al;              // unweighted focal loss

    if (y != 0.0f) {                              // positive
        possum += (x < 0.0f) ? loss * 3.0f : loss; // fn -> * alpha/(1-alpha)
        poscnt += 1;
        return 0u;
    }
    return __float_as_uint(loss) | ((x >= 0.0f) ? 0x80000000u : 0u);
}

__device__ __forceinline__ void hist_one(unsigned key, unsigned pref,
                                         unsigned pm, int lo, unsigned bmk,
                                         unsigned* hA)
{
    unsigned sel = key & 0x7FFFFFFFu;
    if ((sel & pm) == pref)
        atomicAdd(&hA[(sel >> lo) & bmk], 1u);     // ds_add_u32
}

__global__ __launch_bounds__(NT, 1)
void focal_row_kernel(const float* __restrict__ pred,
                      const float* __restrict__ label,
                      float* __restrict__ rowsum,
                      float* __restrict__ rowcnt)
{
    const int tid = threadIdx.x;
    const int row = blockIdx.x;

    extern __shared__ unsigned char smem_raw[];
    unsigned* keys = (unsigned*)smem_raw;          // 65536 u32 = 256 KB
    unsigned* hA   = keys + N_COL;                 // 2048 bins
    unsigned* hB   = hA + NBINS;                   // 2048 bins (scan ping-pong)
    unsigned* misc = hB + NBINS;                   // [0]=npos [1]=prefix/T
                                                   // [2]=krem/quota [3]=bin
                                                   // [4]=above [5]=tie ticket
    float* redS = (float*)(misc + 8);              // 32 wave partial sums
    float* redC = redS + 32;                       // 32 wave partial counts

    if (tid < 8) misc[tid] = 0u;
    __syncthreads();

    const float4* p4 = (const float4*)(pred  + (size_t)row * N_COL);
    const float4* l4 = (const float4*)(label + (size_t)row * N_COL);

    // ---- pass 1: stream HBM once, build keys in LDS, accumulate positives
    float sum   = 0.0f;   // positive contribution (always relevant)
    int   cnt   = 0;
    int   nploc = 0;
    for (int i = tid; i < NV4; i += NT) {
        float4 x = p4[i];                          // global_load_b128
        float4 y = l4[i];
        uint4 kk;
        int pc = 0;
        kk.x = focal_elem(x.x, y.x, sum, pc);
        kk.y = focal_elem(x.y, y.y, sum, pc);
        kk.z = focal_elem(x.z, y.z, sum, pc);
        kk.w = focal_elem(x.w, y.w, sum, pc);
        ((uint4*)keys)[i] = kk;                    // ds_store_b128
        nploc += pc;
        cnt   += pc;
    }
    atomicAdd(&misc[0], (unsigned)nploc);
    __syncthreads();

    const unsigned k = 3u * misc[0];               // floor(3*num_pos) exactly
    if (tid == 0) {
        misc[1] = (k == 0u) ? 0xFFFFFFFFu : 0u;    // T sentinel if k==0
        misc[2] = k;
        misc[3] = 0u; misc[4] = 0u;
    }
    __syncthreads();

    // ---- radix select: k-th largest key (bits 30:0) among negatives
    if (k != 0u) {
        const int      LO [3] = { 21, 10, 0 };
        const unsigned PM [3] = { 0x00000000u, 0x7FE00000u, 0x7FFFFC00u };
        const unsigned BMK[3] = { 1023u, 2047u, 1023u };
        #pragma unroll
        for (int r = 0; r < 3; ++r) {
            for (int b = tid; b < NBINS; b += NT) hA[b] = 0u;
            __syncthreads();
            const unsigned pref = misc[1];
            const unsigned krem = misc[2];
            // vectorized key replay: ds_load_b128, masks constant-folded
            for (int i = tid; i < NV4; i += NT) {
                uint4 kk = ((const uint4*)keys)[i];
                hist_one(kk.x, pref, PM[r], LO[r], BMK[r], hA);
                hist_one(kk.y, pref, PM[r], LO[r], BMK[r], hA);
                hist_one(kk.z, pref, PM[r], LO[r], BMK[r], hA);
                hist_one(kk.w, pref, PM[r], LO[r], BMK[r], hA);
            }
            __syncthreads();
            // suffix sums s[b] = #entries in bins >= b (Hillis-Steele)
            unsigned* src = hA;
            unsigned* dst = hB;
            for (int off = 1; off < NBINS; off <<= 1) {
                for (int b = tid; b < NBINS; b += NT)
                    dst[b] = src[b] + ((b + off < NBINS) ? src[b + off] : 0u);
                __syncthreads();
                unsigned* t = src; src = dst; dst = t;
            }
            // unique crossing bin: s[b] >= krem, s[b+1] < krem
            for (int b = tid; b < NBINS; b += NT) {
                unsigned s  = src[b];
                unsigned sn = (b + 1 < NBINS) ? src[b + 1] : 0u;
                if (s >= krem && sn < krem) { misc[3] = (unsigned)b; misc[4] = sn; }
            }
            __syncthreads();
            if (tid == 0) {
                unsigned c     = misc[3];
                unsigned above = misc[4];
                misc[1] = pref | (c << LO[r]);     // extend threshold prefix
                misc[2] = krem - above;            // still needed among == bin
                misc[3] = 0u; misc[4] = 0u;
            }
            __syncthreads();
        }
    }

    // ---- final pass: LDS keys only (no HBM re-read, no recompute)
    const unsigned T     = misc[1];                // exact threshold key
    const unsigned quota = misc[2];                // ties at T to include
    for (int i = tid; i < NV4; i += NT) {
        uint4 kk = ((const uint4*)keys)[i];        // ds_load_b128
        #pragma unroll
        for (int c4 = 0; c4 < 4; ++c4) {
            unsigned key = (c4 == 0) ? kk.x : (c4 == 1) ? kk.y
                         : (c4 == 2) ? kk.z : kk.w;
            if (key == 0u) continue;               // positive (already counted)
            unsigned sel = key & 0x7FFFFFFFu;
            if (sel > T) {
                sum += __uint_as_float(sel) * ((key >> 31) ? 3.0f : 1.0f);
                cnt += 1;
            } else if (sel == T) {
                unsigned t = atomicAdd(&misc[5], 1u);
                if (t < quota) {
                    sum += __uint_as_float(sel) * ((key >> 31) ? 3.0f : 1.0f);
                    cnt += 1;
                }
            }
        }
    }

    // ---- block reduce (wave32 shfl then cross-wave via LDS)
    float fc = (float)cnt;
    for (int off = 16; off; off >>= 1) {
        sum += __shfl_xor(sum, off, 32);
        fc  += __shfl_xor(fc,  off, 32);
    }
    const int wave = tid >> 5;
    const int lane = tid & 31;
    if (lane == 0) { redS[wave] = sum; redC[wave] = fc; }
    __syncthreads();
    if (wave == 0) {
        float s = redS[lane];
        float c = redC[lane];
        for (int off = 16; off; off >>= 1) {
            s += __shfl_xor(s, off, 32);
            c += __shfl_xor(c, off, 32);
        }
        if (lane == 0) { rowsum[row] = s; rowcnt[row] = c; }
    }
}

// Cross-row reduction with V_WMMA_F32_16X16X4_F32.
// A = all-ones 16x4  =>  D[m][n] = sum_k B[k][n] + C[m][n] (column sums).
// Summing VGPR0 (c[0]) over all 32 lanes yields 2 * total (M=0 and M=8 rows),
// identically for numerator and denominator, so the factor cancels.
__global__ void focal_final_kernel(const float* __restrict__ rs,
                                   const float* __restrict__ rc,
                                   float* __restrict__ out, int nrows)
{
    const int lane = threadIdx.x;                  // one full wave32
    v2f a; a[0] = 1.0f; a[1] = 1.0f;               // ones A-matrix
    v8f cs = {};
    v8f cc = {};
    const int full = nrows & ~63;                  // whole 64-element blocks
    for (int base = 0; base < full; base += 64) {  // unpredicated b64 loads
        float2 s2 = ((const float2*)(rs + base))[lane];
        float2 c2 = ((const float2*)(rc + base))[lane];
        v2f bs; bs[0] = s2.x; bs[1] = s2.y;
        v2f bc; bc[0] = c2.x; bc[1] = c2.y;
        cs = __builtin_amdgcn_wmma_f32_16x16x4_f32(false, a, false, bs,
                                                   (short)0, cs, false, false);
        cc = __builtin_amdgcn_wmma_f32_16x16x4_f32(false, a, false, bc,
                                                   (short)0, cc, false, false);
    }
    if (full < nrows) {                            // tail (never for nrows=128)
        int i0 = full + 2 * lane;
        int i1 = i0 + 1;
        v2f bs, bc;
        bs[0] = (i0 < nrows) ? rs[i0] : 0.0f;
        bs[1] = (i1 < nrows) ? rs[i1] : 0.0f;
        bc[0] = (i0 < nrows) ? rc[i0] : 0.0f;
        bc[1] = (i1 < nrows) ? rc[i1] : 0.0f;
        cs = __builtin_amdgcn_wmma_f32_16x16x4_f32(false, a, false, bs,
                                                   (short)0, cs, false, false);
        cc = __builtin_amdgcn_wmma_f32_16x16x4_f32(false, a, false, bc,
                                                   (short)0, cc, false, false);
    }
    float s = cs[0];
    float c = cc[0];
    for (int off = 16; off; off >>= 1) {
        s += __shfl_xor(s, off, 32);
        c += __shfl_xor(c, off, 32);
    }
    if (lane == 0) out[0] = s / c;                 // (2*sum)/(2*count)
}

extern "C" void kernel_launch(void* const* d_in, const int* in_sizes, int n_in,
                              void* d_out, int out_size, void* d_ws, size_t ws_size,
                              hipStream_t stream) {
    const float* pred  = (const float*)d_in[0];
    const float* label = (const float*)d_in[1];
    float* out = (float*)d_out;

    const int nrows = in_sizes[0] / N_COL;         // 128
    float* rowsum = (float*)d_ws;
    float* rowcnt = rowsum + nrows;

    // 256KB keys + 2x8KB histogram/scan + misc + reduce scratch (< 320KB WGP LDS)
    const size_t smem = (size_t)(N_COL + 2 * NBINS + 8) * sizeof(unsigned)
                      + 64 * sizeof(float);

    focal_row_kernel<<<nrows, NT, smem, stream>>>(pred, label, rowsum, rowcnt);
    focal_final_kernel<<<1, 32, 0, stream>>>(rowsum, rowcnt, out, nrows);
}